// CostConcatenation_87651692577259
// MI455X (gfx1250) — compile-verified
//
#include <hip/hip_runtime.h>
#include <hip/hip_bf16.h>
#include <cstdint>

// Problem constants: left/right (1, 32, 128, 256) f32, disparities -112..15
// (128 values), output (1, 64, 128, 128, 256) f32 = 1 GiB (store-BW bound).
#define C_IN    32
#define D_DISP  128
#define H_DIM   128
#define W_DIM   256
#define MIN_DISP_V (-112)
#define PAD_LO  16                 // idx = w - disp ranges (-16, 368)
#define Q_LEN   (W_DIM + 128)      // 384 slots: [0,16) zero, [16,272) data, [272,384) zero

typedef float v4f __attribute__((ext_vector_type(4)));

#if defined(__AMDGCN__) && __has_builtin(__builtin_amdgcn_global_load_async_to_lds_b128)
#define USE_ASYNC_LDS 1
typedef int v4i __attribute__((vector_size(16)));
typedef __attribute__((address_space(1))) v4i global_v4i;
typedef __attribute__((address_space(3))) v4i lds_v4i;
#else
#define USE_ASYNC_LDS 0
#endif

__global__ __launch_bounds__(256) void CostConcat_kernel(const float* __restrict__ left,
                                                         const float* __restrict__ right,
                                                         float* __restrict__ out) {
    __shared__ float rowbuf[W_DIM];   // staged source row
    __shared__ float Q[Q_LEN];        // zero-padded row (right) or 0/1 mask (left)

    const int tid = threadIdx.x;
    const int bid = blockIdx.x;            // bid = c * H + h, c in [0,64)
    const int h   = bid & (H_DIM - 1);
    const int c   = bid >> 7;
    const bool isLeft = (c < C_IN);

    const float* src = isLeft
        ? (left  + ((size_t)c          * H_DIM + h) * W_DIM)
        : (right + ((size_t)(c - C_IN) * H_DIM + h) * W_DIM);

    // ---- Stage the 1 KiB source row into LDS via CDNA5 async global->LDS ----
#if USE_ASYNC_LDS
    if (tid < (W_DIM / 4)) {
        global_v4i* gsrc = (global_v4i*)(uint64_t)(uintptr_t)(src + tid * 4);
        lds_v4i*    ldst = (lds_v4i*)(uint32_t)(uintptr_t)(&rowbuf[tid * 4]);
        __builtin_amdgcn_global_load_async_to_lds_b128(gsrc, ldst, 0, 0);
    }
#if __has_builtin(__builtin_amdgcn_s_wait_asynccnt)
    __builtin_amdgcn_s_wait_asynccnt(0);
#else
    asm volatile("s_wait_asynccnt 0" ::: "memory");
#endif
#else
    if (tid < W_DIM) rowbuf[tid] = src[tid];
#endif
    __syncthreads();

    // ---- Build padded Q: mid = data (right) or 1.0 (left); pads = 0 ----
#pragma unroll
    for (int i = tid; i < Q_LEN; i += 256) {
        const bool mid = (i >= PAD_LO) && (i < PAD_LO + W_DIM);
        Q[i] = mid ? (isLeft ? 1.0f : rowbuf[i - PAD_LO]) : 0.0f;
    }

    // Per-lane scale: left -> its 4 row values; right -> 1.0
    const int w0   = (tid & 63) * 4;
    const int dsub = tid >> 6;
    float S[4];
#pragma unroll
    for (int k = 0; k < 4; ++k) S[k] = isLeft ? rowbuf[w0 + k] : 1.0f;

    __syncthreads();

    // ---- Stream 128 shifted/masked rows: 1 ds_load + 1 v_mul per element ----
    // Qi(d, w) = w - disp + PAD_LO = (w0 + 128 - dsub) - dbase + k  (always in range)
    const int qbase = w0 + 128 - dsub;
    const size_t out_base = (size_t)c * D_DISP * H_DIM * W_DIM + (size_t)h * W_DIM
                          + (size_t)dsub * (H_DIM * W_DIM) + w0;

    for (int dbase = 0; dbase < D_DISP; dbase += 4) {
        v4f v;
#pragma unroll
        for (int k = 0; k < 4; ++k) {
            v[k] = Q[qbase - dbase + k] * S[k];
        }
        float* dst = out + out_base + (size_t)dbase * (H_DIM * W_DIM);
        __builtin_nontemporal_store(v, (v4f*)dst);   // 16B NT streaming store
    }
}

extern "C" void kernel_launch(void* const* d_in, const int* in_sizes, int n_in,
                              void* d_out, int out_size, void* d_ws, size_t ws_size,
                              hipStream_t stream) {
    const float* left  = (const float*)d_in[0];
    const float* right = (const float*)d_in[1];
    float* out = (float*)d_out;

    dim3 grid(64 * H_DIM);   // one block per (c, h): 8192 blocks
    dim3 block(256);         // 8 wave32 waves
    hipLaunchKernelGGL(CostConcat_kernel, grid, block, 0, stream, left, right, out);
}